// QuantileLoss_50491635532156
// MI455X (gfx1250) — compile-verified
//
#include <hip/hip_runtime.h>

typedef float v2f __attribute__((ext_vector_type(2)));
typedef float v4f __attribute__((ext_vector_type(4)));
typedef float v8f __attribute__((ext_vector_type(8)));

// ---------------------------------------------------------------------------
// Kernel 1: bandwidth-bound partial reduction.
// Each "group" = 4 rows: preds 20 floats (5 x b128), target 12 floats (3 x b128),
// both 16B-aligned for every group. Non-temporal: 268 MB streamed once.
// ---------------------------------------------------------------------------
__global__ __launch_bounds__(256) void ql_partials(
    const float* __restrict__ preds, const float* __restrict__ target,
    float* __restrict__ partials, long long ngroups)
{
    const v4f* __restrict__ p4 = reinterpret_cast<const v4f*>(preds);
    const v4f* __restrict__ t4 = reinterpret_cast<const v4f*>(target);

    float acc = 0.0f;
    const long long stride = (long long)gridDim.x * blockDim.x;
    for (long long g = (long long)blockIdx.x * blockDim.x + threadIdx.x;
         g < ngroups; g += stride) {
        v4f pv[5], tv[3];
#pragma unroll
        for (int i = 0; i < 5; ++i) pv[i] = __builtin_nontemporal_load(&p4[5 * g + i]);
#pragma unroll
        for (int i = 0; i < 3; ++i) tv[i] = __builtin_nontemporal_load(&t4[3 * g + i]);

        float P[20], T[12];
#pragma unroll
        for (int i = 0; i < 5; ++i) {
            P[4 * i + 0] = pv[i][0]; P[4 * i + 1] = pv[i][1];
            P[4 * i + 2] = pv[i][2]; P[4 * i + 3] = pv[i][3];
        }
#pragma unroll
        for (int i = 0; i < 3; ++i) {
            T[4 * i + 0] = tv[i][0]; T[4 * i + 1] = tv[i][1];
            T[4 * i + 2] = tv[i][2]; T[4 * i + 3] = tv[i][3];
        }

#pragma unroll
        for (int j = 0; j < 4; ++j) {
            float d0 = P[5 * j + 0] - T[3 * j + 0];
            float d1 = P[5 * j + 1] - T[3 * j + 1];
            float d2 = P[5 * j + 2] - T[3 * j + 2];
            acc = fmaf(d0, d0, acc);
            acc = fmaf(d1, d1, acc);
            acc = fmaf(d2, d2, acc);
            float p2  = P[5 * j + 2];
            float p3  = P[5 * j + 3];
            float t2s = T[3 * j + 2] * 0.95f;
            float df  = p3 - t2s;
            float lower = (p3 > p2) ? 1000.0f : ((p3 > t2s) ? 0.0f : df * df);
            acc += 2.0f * lower;   // lower + upper
        }
    }

    // wave32 butterfly reduction
#pragma unroll
    for (int off = 16; off > 0; off >>= 1) acc += __shfl_xor(acc, off, 32);

    __shared__ float wsum[8];
    const int lane = threadIdx.x & 31;
    const int wave = threadIdx.x >> 5;
    if (lane == 0) wsum[wave] = acc;
    __syncthreads();
    if (threadIdx.x == 0) {
        float s = 0.0f;
#pragma unroll
        for (int i = 0; i < 8; ++i) s += wsum[i];
        partials[blockIdx.x] = s;
    }
}

// ---------------------------------------------------------------------------
// Kernel 2: single wave32. Reduce nParts partials (multiple of 64) with
// V_WMMA_F32_16X16X4_F32: A = ones(16x4), B = 64 partials per instruction.
// With A all-ones, sum over all D entries of one row == sum of all B elements,
// independent of B's lane layout. Sum of c[0] over 32 lanes == 2 * total.
// ---------------------------------------------------------------------------
__global__ __launch_bounds__(32) void ql_final(
    const float* __restrict__ partials, int nParts,
    const float* __restrict__ preds, const float* __restrict__ target,
    long long tailStart, long long nRows, float invCount,
    float* __restrict__ out)
{
    const int lane = threadIdx.x;       // 32 lanes, no divergence -> EXEC all 1s
    v2f a; a[0] = 1.0f; a[1] = 1.0f;    // all-ones A matrix
    v8f c = {};

    for (int base = 0; base < nParts; base += 64) {
        v2f b;
        b[0] = partials[base + 2 * lane + 0];
        b[1] = partials[base + 2 * lane + 1];
        c = __builtin_amdgcn_wmma_f32_16x16x4_f32(
                /*neg_a=*/false, a, /*neg_b=*/false, b,
                /*c_mod=*/(short)0, c, /*reuse_a=*/false, /*reuse_b=*/false);
    }

    float s = c[0];
#pragma unroll
    for (int off = 16; off > 0; off >>= 1) s += __shfl_xor(s, off, 32);
    float total = 0.5f * s;             // rows 0 and 8 both counted in VGPR0

    if (lane == 0) {
        for (long long r = tailStart; r < nRows; ++r) {   // generic N%4 tail
            float d0 = preds[5 * r + 0] - target[3 * r + 0];
            float d1 = preds[5 * r + 1] - target[3 * r + 1];
            float d2 = preds[5 * r + 2] - target[3 * r + 2];
            total += d0 * d0 + d1 * d1 + d2 * d2;
            float p2  = preds[5 * r + 2];
            float p3  = preds[5 * r + 3];
            float t2s = target[3 * r + 2] * 0.95f;
            float df  = p3 - t2s;
            float lower = (p3 > p2) ? 1000.0f : ((p3 > t2s) ? 0.0f : df * df);
            total += 2.0f * lower;
        }
        out[0] = total * invCount;
    }
}

extern "C" void kernel_launch(void* const* d_in, const int* in_sizes, int n_in,
                              void* d_out, int out_size, void* d_ws, size_t ws_size,
                              hipStream_t stream) {
    const float* preds  = (const float*)d_in[0];   // (N,5) f32
    const float* target = (const float*)d_in[1];   // (N,3) f32
    const long long N = (long long)in_sizes[0] / 5;
    const long long G = N / 4;
    const long long tailStart = G * 4;

    int blocks = 4096;                              // 16 KB of partials
    if ((size_t)blocks * sizeof(float) > ws_size)
        blocks = (int)(ws_size / sizeof(float));
    blocks &= ~63;                                  // multiple of 64 for WMMA chunks
    if (blocks < 64) blocks = 64;

    float* parts = (float*)d_ws;
    ql_partials<<<blocks, 256, 0, stream>>>(preds, target, parts, G);

    const float invCount = 1.0f / (5.0f * (float)N);
    ql_final<<<1, 32, 0, stream>>>(parts, blocks, preds, target,
                                   tailStart, N, invCount, (float*)d_out);
}